// PointTransformer_15710990369321
// MI455X (gfx1250) — compile-verified
//
#include <hip/hip_runtime.h>
#include <hip/hip_bf16.h>
#include <cmath>

// ---------------- CDNA5 fp32 WMMA types ----------------
typedef __attribute__((ext_vector_type(2))) float v2f;
typedef __attribute__((ext_vector_type(8))) float v8f;

#define BATCH 4
#define NPTS  2048
#define DM    512
#define NNB   16     // nsample / knn k
#define GCLS  8      // global cls points
#define CHUNK 512    // transformer chunk over query points

static inline unsigned nb(size_t total, int bs) {
  return (unsigned)((total + (size_t)bs - 1) / (size_t)bs);
}

// =====================================================================
// FAST GEMM (K%16==0, M%16==0, N%16==0): C = A@W (+bias)(+relu)
// 128 threads = 4 waves. All 4 waves share one 16-row A tile staged in
// LDS (padded stride 18 -> conflict-free, 8B-aligned pair reads); each
// wave owns a 16-col tile of N. 4 fp32 WMMAs per LDS tile.
// =====================================================================
#define BKF 16
template <int HASB, int RELU>
__global__ __launch_bounds__(128) void gemm_fast(const float* __restrict__ A,
                                                 const float* __restrict__ W,
                                                 const float* __restrict__ bias,
                                                 float* __restrict__ C,
                                                 int M, int K, int N) {
  __shared__ float As[16][BKF + 2];   // stride 18 floats
  const int tid  = threadIdx.x;
  const int lane = tid & 31;
  const int wid  = tid >> 5;
  const int half = lane >> 4;         // 0: K 0/1, 1: K 2/3 within a step
  const int l15  = lane & 15;
  const int tileM = blockIdx.x * 16;                 // always < M (M%16==0)
  const int tileN = blockIdx.y * 64 + wid * 16;      // may exceed N
  const int n     = tileN + l15;
  const int nld   = (n < N) ? n : (N - 1);           // clamped load col

  // cooperative A-tile load mapping: 128 threads cover 16x16 in 2 passes
  const int ar = tid >> 4;            // 0..7
  const int ac = tid & 15;            // 0..15
  const float* Ag0 = A + (size_t)(tileM + ar) * K + ac;
  const float* Ag1 = A + (size_t)(tileM + ar + 8) * K + ac;

  v8f acc = {};
  for (int k0 = 0; k0 < K; k0 += BKF) {
    __syncthreads();                  // previous tile fully consumed
    As[ar][ac]     = Ag0[k0];
    As[ar + 8][ac] = Ag1[k0];
    __syncthreads();
#pragma unroll
    for (int kk = 0; kk < BKF; kk += 4) {
      v2f a, b;
      a.x = As[l15][kk + half * 2];
      a.y = As[l15][kk + half * 2 + 1];
      const float* bp = W + (size_t)(k0 + kk + half * 2) * N + nld;
      b.x = bp[0];
      b.y = bp[N];
      acc = __builtin_amdgcn_wmma_f32_16x16x4_f32(false, a, false, b,
                                                  (short)0, acc, false, false);
    }
  }
  if (n < N) {
    const float bv = HASB ? bias[n] : 0.f;
#pragma unroll
    for (int r = 0; r < 8; ++r) {
      const int row = tileM + r + half * 8;
      float v = acc[r] + bv;
      if (RELU) v = fmaxf(v, 0.f);
      C[(size_t)row * N + n] = v;
    }
  }
}

// =====================================================================
// GENERIC GEMM (any M,K,N): clamped loads + select-to-zero (no exec-mask
// branch storm), one fp32 WMMA per K-step of 4.
// =====================================================================
template <int HASB, int RELU>
__global__ __launch_bounds__(128) void gemm_gen(const float* __restrict__ A,
                                                const float* __restrict__ W,
                                                const float* __restrict__ bias,
                                                float* __restrict__ C,
                                                int M, int K, int N) {
  const int lane = threadIdx.x & 31;
  const int wid  = threadIdx.x >> 5;
  const int half = lane >> 4;
  const int l15  = lane & 15;
  const int tileM = blockIdx.x * 16;
  const int tileN = (blockIdx.y * 4 + wid) * 16;
  if (tileN >= N) return;             // wave-uniform

  const int m  = tileM + l15;
  const int n  = tileN + l15;
  const int mc = (m < M) ? m : (M - 1);
  const int nc = (n < N) ? n : (N - 1);
  v8f acc = {};
  for (int k0 = 0; k0 < K; k0 += 4) {
    const int ka  = k0 + half * 2;
    const int k0c = (ka < K) ? ka : (K - 1);
    const int k1c = (ka + 1 < K) ? (ka + 1) : (K - 1);
    const float ax = A[(size_t)mc * K + k0c];
    const float ay = A[(size_t)mc * K + k1c];
    const float bx = W[(size_t)k0c * N + nc];
    const float by = W[(size_t)k1c * N + nc];
    v2f a, b;
    a.x = (m < M && ka < K)     ? ax : 0.f;
    a.y = (m < M && ka + 1 < K) ? ay : 0.f;
    b.x = (n < N && ka < K)     ? bx : 0.f;
    b.y = (n < N && ka + 1 < K) ? by : 0.f;
    acc = __builtin_amdgcn_wmma_f32_16x16x4_f32(false, a, false, b,
                                                (short)0, acc, false, false);
  }
  if (n < N) {
    const float bv = HASB ? bias[n] : 0.f;
#pragma unroll
    for (int r = 0; r < 8; ++r) {
      const int row = tileM + r + half * 8;
      if (row < M) {
        float v = acc[r] + bv;
        if (RELU) v = fmaxf(v, 0.f);
        C[(size_t)row * N + n] = v;
      }
    }
  }
}

static void gemm(const float* A, const float* W, const float* bias, float* C,
                 int M, int K, int N, bool relu, hipStream_t s) {
  dim3 blk(128);
  if ((K & 15) == 0 && (M & 15) == 0 && (N & 15) == 0) {
    dim3 grid(nb(M, 16), nb(N, 64));
    if (bias) {
      if (relu) gemm_fast<1, 1><<<grid, blk, 0, s>>>(A, W, bias, C, M, K, N);
      else      gemm_fast<1, 0><<<grid, blk, 0, s>>>(A, W, bias, C, M, K, N);
    } else {
      if (relu) gemm_fast<0, 1><<<grid, blk, 0, s>>>(A, W, bias, C, M, K, N);
      else      gemm_fast<0, 0><<<grid, blk, 0, s>>>(A, W, bias, C, M, K, N);
    }
  } else {
    dim3 grid(nb(M, 16), nb(N, 64));
    if (bias) {
      if (relu) gemm_gen<1, 1><<<grid, blk, 0, s>>>(A, W, bias, C, M, K, N);
      else      gemm_gen<1, 0><<<grid, blk, 0, s>>>(A, W, bias, C, M, K, N);
    } else {
      if (relu) gemm_gen<0, 1><<<grid, blk, 0, s>>>(A, W, bias, C, M, K, N);
      else      gemm_gen<0, 0><<<grid, blk, 0, s>>>(A, W, bias, C, M, K, N);
    }
  }
}

// =====================================================================
// Elementwise / irregular kernels
// =====================================================================
__global__ void k_slice_xyz(const float* __restrict__ x, float* __restrict__ xyz,
                            size_t total) {
  size_t e = (size_t)blockIdx.x * blockDim.x + threadIdx.x;
  if (e >= total) return;
  size_t r = e / 3; int c = (int)(e % 3);
  xyz[e] = x[r * 6 + c];
}

// KNN: for each query, 16 nearest points of db (stable, ascending distance),
// then optionally append global indices Nd-g .. Nd-1.
__global__ void k_knn(const float* __restrict__ Q, const float* __restrict__ D,
                      int* __restrict__ out, int B_, int Nq, int Nd, int kt, int g) {
  int t = blockIdx.x * blockDim.x + threadIdx.x;
  if (t >= B_ * Nq) return;
  int b = t / Nq, q = t % Nq;
  const float* qp = Q + ((size_t)b * Nq + q) * 3;
  const float qx = qp[0], qy = qp[1], qz = qp[2];
  const float* dp = D + (size_t)b * Nd * 3;
  float bd[NNB]; int bi[NNB];
#pragma unroll
  for (int i = 0; i < NNB; ++i) { bd[i] = 3.4e38f; bi[i] = 0; }
  for (int n = 0; n < Nd; ++n) {
    float dx = dp[n * 3 + 0] - qx;
    float dy = dp[n * 3 + 1] - qy;
    float dz = dp[n * 3 + 2] - qz;
    float d = dx * dx + dy * dy + dz * dz;
    if (d < bd[NNB - 1]) {           // strict: stable tie-break like lax.top_k
      int j = NNB - 1;
      while (j > 0 && bd[j - 1] > d) { bd[j] = bd[j - 1]; bi[j] = bi[j - 1]; --j; }
      bd[j] = d; bi[j] = n;
    }
  }
  int* o = out + (size_t)t * kt;
#pragma unroll
  for (int i = 0; i < NNB; ++i) o[i] = bi[i];
  for (int gi = 0; gi < kt - NNB; ++gi) o[NNB + gi] = Nd - g + gi;
}

// Farthest point sampling: one block per batch. Matches jax.lax.scan semantics:
// record current 'far' first, then update distances and take argmax (first max).
__global__ void k_fps(const float* __restrict__ xyz, int* __restrict__ out,
                      int N, int npoint) {
  __shared__ float dist[2048];
  __shared__ float rv[256];
  __shared__ int   ri[256];
  __shared__ int   s_far;
  const int b = blockIdx.x, t = threadIdx.x;
  const float* P = xyz + (size_t)b * N * 3;
  for (int n = t; n < N; n += blockDim.x) dist[n] = 1e10f;
  if (t == 0) s_far = 0;
  __syncthreads();
  for (int it = 0; it < npoint; ++it) {
    const int far = s_far;
    if (t == 0) out[(size_t)b * npoint + it] = far;
    const float cx = P[far * 3 + 0], cy = P[far * 3 + 1], cz = P[far * 3 + 2];
    float bv = -1.f; int bidx = 0;
    for (int n = t; n < N; n += blockDim.x) {
      float dx = P[n * 3 + 0] - cx;
      float dy = P[n * 3 + 1] - cy;
      float dz = P[n * 3 + 2] - cz;
      float d = dx * dx + dy * dy + dz * dz;
      float nd = fminf(dist[n], d);
      dist[n] = nd;
      if (nd > bv) { bv = nd; bidx = n; }   // strict -> first max in stripe
    }
    rv[t] = bv; ri[t] = bidx;
    __syncthreads();
    for (int sft = 128; sft > 0; sft >>= 1) {
      if (t < sft) {
        if (rv[t + sft] > rv[t] || (rv[t + sft] == rv[t] && ri[t + sft] < ri[t])) {
          rv[t] = rv[t + sft]; ri[t] = ri[t + sft];
        }
      }
      __syncthreads();
    }
    if (t == 0) s_far = ri[0];
    __syncthreads();
  }
}

// Generic per-batch row gather: dst[r,c] = src[(b*Nd + idx[r])*C + c]
__global__ void k_gather(const float* __restrict__ src, const int* __restrict__ idx,
                         float* __restrict__ dst, int rows_per_b, int Nd, int C,
                         size_t total) {
  size_t e = (size_t)blockIdx.x * blockDim.x + threadIdx.x;
  if (e >= total) return;
  size_t r = e / C; int c = (int)(e % C);
  int b = (int)(r / rows_per_b);
  dst[e] = src[((size_t)b * Nd + idx[r]) * C + c];
}

// Grouped features for set_abstraction:
// dst[b,i,j,0:3] = xyz[idx]-new_xyz[b,i];  dst[b,i,j,3:3+Cin] = pts[idx]
__global__ void k_group(const float* __restrict__ xyz, const float* __restrict__ pts,
                        const int* __restrict__ idx, const float* __restrict__ nxyz,
                        float* __restrict__ dst, int Np, int S, int Nd, int Cin,
                        size_t total) {
  size_t e = (size_t)blockIdx.x * blockDim.x + threadIdx.x;
  if (e >= total) return;
  const int Ct = 3 + Cin;
  int c = (int)(e % Ct); size_t r = e / Ct;
  size_t bi = r / S;
  int i = (int)(bi % Np); int b = (int)(bi / Np);
  int id = idx[r];
  if (c < 3)
    dst[e] = xyz[((size_t)b * Nd + id) * 3 + c] - nxyz[((size_t)b * Np + i) * 3 + c];
  else
    dst[e] = pts[((size_t)b * Nd + id) * Cin + (c - 3)];
}

// pos-MLP input (chunked): Dd[b,n',j,c] = xyz[b,n]-xyz[b,knn[b,n,j]], n=n0+n'
__global__ void k_delta(const float* __restrict__ xyz, const int* __restrict__ knn,
                        float* __restrict__ Dd, int N, int n0, int Nc, int kt,
                        size_t total) {
  size_t e = (size_t)blockIdx.x * blockDim.x + threadIdx.x;
  if (e >= total) return;
  int c = (int)(e % 3); size_t r = e / 3;
  int j = (int)(r % kt); size_t bn = r / kt;
  int np = (int)(bn % Nc); int b = (int)(bn / Nc);
  int n = n0 + np;
  int id = knn[((size_t)b * N + n) * kt + j];
  Dd[e] = xyz[((size_t)b * N + n) * 3 + c] - xyz[((size_t)b * N + id) * 3 + c];
}

// attn input (chunked): A = q[b,n] - k[b,knn] + pos
__global__ void k_qkp(const float* __restrict__ Qb, const float* __restrict__ Kb,
                      const float* __restrict__ P, const int* __restrict__ knn,
                      float* __restrict__ A, int N, int n0, int Nc, int kt, int F,
                      size_t total) {
  size_t e = (size_t)blockIdx.x * blockDim.x + threadIdx.x;
  if (e >= total) return;
  int f = (int)(e % F); size_t r = e / F;
  int j = (int)(r % kt); size_t bn = r / kt;
  int np = (int)(bn % Nc); int b = (int)(bn / Nc);
  int n = n0 + np;
  int id = knn[((size_t)b * N + n) * kt + j];
  A[e] = Qb[((size_t)b * N + n) * F + f] - Kb[((size_t)b * N + id) * F + f] + P[e];
}

// softmax over neighbor axis (axis=-2) per (b,n,f), with scale 1/sqrt(DM)
__global__ void k_softmax(float* __restrict__ A, int BNc, int kt, int F, float scale,
                          size_t total) {
  size_t t = (size_t)blockIdx.x * blockDim.x + threadIdx.x;
  if (t >= total) return;
  size_t bn = t / F; int f = (int)(t % F);
  float* p = A + ((size_t)bn * kt) * F + f;
  float mx = -3.4e38f;
  for (int j = 0; j < kt; ++j) mx = fmaxf(mx, p[(size_t)j * F] * scale);
  float sum = 0.f;
  for (int j = 0; j < kt; ++j) {
    float ev = expf(p[(size_t)j * F] * scale - mx);
    p[(size_t)j * F] = ev;
    sum += ev;
  }
  float inv = 1.f / sum;
  for (int j = 0; j < kt; ++j) p[(size_t)j * F] *= inv;
  (void)BNc;
}

// res[b,n,f] = sum_j attn[b,n,j,f] * (v[b,knn,f] + pos[b,n,j,f])   (chunked)
__global__ void k_attnv(const float* __restrict__ A, const float* __restrict__ Vb,
                        const float* __restrict__ P, const int* __restrict__ knn,
                        float* __restrict__ R, int N, int n0, int Nc, int kt, int F,
                        size_t total) {
  size_t t = (size_t)blockIdx.x * blockDim.x + threadIdx.x;
  if (t >= total) return;
  int f = (int)(t % F); size_t bn = t / F;
  int np = (int)(bn % Nc); int b = (int)(bn / Nc);
  int n = n0 + np;
  float acc = 0.f;
  for (int j = 0; j < kt; ++j) {
    size_t r = bn * kt + j;
    int id = knn[((size_t)b * N + n) * kt + j];
    acc += A[r * F + f] * (Vb[((size_t)b * N + id) * F + f] + P[r * F + f]);
  }
  R[t] = acc;
}

// scatter chunk rows back + residual: dst[b,n0+r,:] = src[b,r,:] + resid[b,n0+r,:]
__global__ void k_scatter_add(const float* __restrict__ src, const float* __restrict__ resid,
                              float* __restrict__ dst, int N, int n0, int Nc, int C,
                              size_t total) {
  size_t e = (size_t)blockIdx.x * blockDim.x + threadIdx.x;
  if (e >= total) return;
  int c = (int)(e % C); size_t r = e / C;
  int np = (int)(r % Nc); int b = (int)(r / Nc);
  size_t di = ((size_t)b * N + n0 + np) * C + c;
  dst[di] = src[e] + resid[di];
}

// max over S samples: out[bi,f] = max_j in[(bi*S+j)*C+f]
__global__ void k_maxpool(const float* __restrict__ in, float* __restrict__ out,
                          int S, int C, size_t total) {
  size_t t = (size_t)blockIdx.x * blockDim.x + threadIdx.x;
  if (t >= total) return;
  size_t bi = t / C; int f = (int)(t % C);
  float m = -3.4e38f;
  for (int j = 0; j < S; ++j) m = fmaxf(m, in[((size_t)bi * S + j) * C + f]);
  out[t] = m;
}

// blocked 2D copy with per-batch leading dims
__global__ void k_copy2d(const float* __restrict__ src, float* __restrict__ dst,
                         int rows, int C, int src_ld, int dst_ld,
                         int src_r0, int dst_r0, size_t total) {
  size_t e = (size_t)blockIdx.x * blockDim.x + threadIdx.x;
  if (e >= total) return;
  int c = (int)(e % C); size_t r = e / C;
  int rr = (int)(r % rows); int b = (int)(r / rows);
  dst[((size_t)b * dst_ld + dst_r0 + rr) * C + c] =
      src[((size_t)b * src_ld + src_r0 + rr) * C + c];
}

__global__ void k_mean(const float* __restrict__ pts, float* __restrict__ out,
                       int N, int C, size_t total) {
  size_t t = (size_t)blockIdx.x * blockDim.x + threadIdx.x;
  if (t >= total) return;
  int f = (int)(t % C); int b = (int)(t / C);
  float s = 0.f;
  for (int n = 0; n < N; ++n) s += pts[((size_t)b * N + n) * C + f];
  out[t] = s / (float)N;
}

// =====================================================================
// Host-side parameter walking (setup_inputs dict-insertion order, DFS)
// =====================================================================
struct Reader { void* const* in; int cur;
  const float* next() { return (const float*)in[cur++]; } };
struct Lin  { const float* w; const float* b; };
struct Mlp2 { Lin l1, l2; };
struct TF   { Lin fc1; const float* wq; const float* wk; const float* wv;
              Mlp2 fc_delta, fc_gamma; Lin fc2; };
struct TD   { Lin sa1, sa2; Mlp2 cls; };
struct Block{ TD td; TF tf; };

static Lin  rd_lin (Reader& r) { Lin l; l.w = r.next(); l.b = r.next(); return l; }
static Mlp2 rd_mlp2(Reader& r) { Mlp2 m; m.l1 = rd_lin(r); m.l2 = rd_lin(r); return m; }
static TF   rd_tf  (Reader& r) {
  TF t; t.fc1 = rd_lin(r); t.wq = r.next(); t.wk = r.next(); t.wv = r.next();
  t.fc_delta = rd_mlp2(r); t.fc_gamma = rd_mlp2(r); t.fc2 = rd_lin(r); return t;
}

struct WS {
  float *X, *Q, *K, *V, *P, *H, *A, *R, *T, *Dd;
  float *NX, *NP, *G, *H1, *H2, *CLSI, *CLSH, *CLSO;
  float *xyzA, *xyzB, *ptsA, *ptsB, *MEAN, *HD1, *HD2;
  int *knn, *fpsidx;
};

// =====================================================================
// Transformer (chunked over query points)
// =====================================================================
static void run_tf(const TF& p, const float* xyz, const float* fin, float* fout,
                   int N, int dpts, int k, int g, WS& w, hipStream_t s) {
  const int kt = k + g;
  const int BN = BATCH * N;
  const float scale = 1.0f / sqrtf((float)DM);

  k_knn<<<nb((size_t)BN, 128), 128, 0, s>>>(xyz, xyz, w.knn, BATCH, N, N, kt, g);
  gemm(fin, p.fc1.w, p.fc1.b, w.X, BN, dpts, DM, false, s);
  gemm(w.X, p.wq, nullptr, w.Q, BN, DM, DM, false, s);
  gemm(w.X, p.wk, nullptr, w.K, BN, DM, DM, false, s);
  gemm(w.X, p.wv, nullptr, w.V, BN, DM, DM, false, s);

  for (int n0 = 0; n0 < N; n0 += CHUNK) {
    const int Nc = (N - n0 < CHUNK) ? (N - n0) : CHUNK;
    const int rows = BATCH * Nc * kt;
    size_t t3 = (size_t)rows * 3, tF = (size_t)rows * DM;
    size_t tR = (size_t)BATCH * Nc * DM, tO = (size_t)BATCH * Nc * dpts;

    k_delta<<<nb(t3, 256), 256, 0, s>>>(xyz, w.knn, w.Dd, N, n0, Nc, kt, t3);
    gemm(w.Dd, p.fc_delta.l1.w, p.fc_delta.l1.b, w.H, rows, 3, DM, true, s);
    gemm(w.H, p.fc_delta.l2.w, p.fc_delta.l2.b, w.P, rows, DM, DM, false, s);
    k_qkp<<<nb(tF, 256), 256, 0, s>>>(w.Q, w.K, w.P, w.knn, w.A, N, n0, Nc, kt, DM, tF);
    gemm(w.A, p.fc_gamma.l1.w, p.fc_gamma.l1.b, w.H, rows, DM, DM, true, s);
    gemm(w.H, p.fc_gamma.l2.w, p.fc_gamma.l2.b, w.A, rows, DM, DM, false, s);
    k_softmax<<<nb(tR, 256), 256, 0, s>>>(w.A, BATCH * Nc, kt, DM, scale, tR);
    k_attnv<<<nb(tR, 256), 256, 0, s>>>(w.A, w.V, w.P, w.knn, w.R, N, n0, Nc, kt, DM, tR);
    gemm(w.R, p.fc2.w, p.fc2.b, w.T, BATCH * Nc, DM, dpts, false, s);
    k_scatter_add<<<nb(tO, 256), 256, 0, s>>>(w.T, fin, fout, N, n0, Nc, dpts, tO);
  }
}

// =====================================================================
// Set abstraction: fps -> knn group -> 2x (lin+relu) -> maxpool
// =====================================================================
static void run_sa(const TD& td, const float* xyz, const float* pts,
                   int N, int Cin, int Np, int ch, WS& w, hipStream_t s) {
  k_fps<<<dim3(BATCH), 256, 0, s>>>(xyz, w.fpsidx, N, Np);
  {
    size_t t = (size_t)BATCH * Np * 3;
    k_gather<<<nb(t, 256), 256, 0, s>>>(xyz, w.fpsidx, w.NX, Np, N, 3, t);
  }
  k_knn<<<nb((size_t)BATCH * Np, 128), 128, 0, s>>>(w.NX, xyz, w.knn, BATCH, Np, N, NNB, 0);
  {
    size_t t = (size_t)BATCH * Np * NNB * (3 + Cin);
    k_group<<<nb(t, 256), 256, 0, s>>>(xyz, pts, w.knn, w.NX, w.G, Np, NNB, N, Cin, t);
  }
  const int rows = BATCH * Np * NNB;
  gemm(w.G,  td.sa1.w, td.sa1.b, w.H1, rows, 3 + Cin, ch, true, s);
  gemm(w.H1, td.sa2.w, td.sa2.b, w.H2, rows, ch, ch, true, s);
  {
    size_t t = (size_t)BATCH * Np * ch;
    k_maxpool<<<nb(t, 256), 256, 0, s>>>(w.H2, w.NP, NNB, ch, t);
  }
}

// =====================================================================
// Entry point
// =====================================================================
extern "C" void kernel_launch(void* const* d_in, const int* in_sizes, int n_in,
                              void* d_out, int out_size, void* d_ws, size_t ws_size,
                              hipStream_t stream) {
  (void)in_sizes; (void)n_in; (void)out_size; (void)ws_size;

  // ---- parameters (dict insertion order, depth-first) ----
  Reader rd{d_in, 0};
  const float* x = rd.next();           // (B, 2048, 6)
  Mlp2 fc1 = rd_mlp2(rd);
  TF t1 = rd_tf(rd);
  Block blk[4];
  for (int i = 0; i < 4; ++i) {
    blk[i].td.sa1 = rd_lin(rd);
    blk[i].td.sa2 = rd_lin(rd);
    blk[i].td.cls = rd_mlp2(rd);
    blk[i].tf = rd_tf(rd);
  }
  Lin hl1 = rd_lin(rd), hl2 = rd_lin(rd), hl3 = rd_lin(rd);

  // ---- workspace carve (bump allocator, ~400 MB peak) ----
  char* wp = (char*)d_ws;
  auto alloc = [&](size_t bytes) -> void* {
    void* p = (void*)wp;
    wp += (bytes + 255) & ~(size_t)255;
    return p;
  };
  WS w;
  const size_t BNmax = (size_t)BATCH * NPTS;                 // 8192
  const size_t CROWS = (size_t)BATCH * CHUNK * (NNB + GCLS); // max chunk (n,k) rows
  w.X  = (float*)alloc(BNmax * DM * 4);
  w.Q  = (float*)alloc(BNmax * DM * 4);
  w.K  = (float*)alloc(BNmax * DM * 4);
  w.V  = (float*)alloc(BNmax * DM * 4);
  w.P  = (float*)alloc(CROWS * DM * 4);
  w.H  = (float*)alloc(CROWS * DM * 4);
  w.A  = (float*)alloc(CROWS * DM * 4);
  w.R  = (float*)alloc((size_t)BATCH * CHUNK * DM * 4);
  w.T  = (float*)alloc((size_t)BATCH * CHUNK * DM * 4);
  w.Dd = (float*)alloc(CROWS * 3 * 4);
  w.NX = (float*)alloc((size_t)BATCH * 512 * 3 * 4);
  w.NP = (float*)alloc((size_t)BATCH * 512 * 64 * 4);        // max Np*ch = 512*64
  w.G  = (float*)alloc((size_t)BATCH * 512 * NNB * 35 * 4);  // max grouped
  w.H1 = (float*)alloc((size_t)BATCH * 512 * NNB * 64 * 4);  // max rows*ch
  w.H2 = (float*)alloc((size_t)BATCH * 512 * NNB * 64 * 4);
  w.CLSI = (float*)alloc((size_t)BATCH * 8 * 512 * 4);
  w.CLSH = (float*)alloc((size_t)BATCH * 8 * 512 * 4);
  w.CLSO = (float*)alloc((size_t)BATCH * 8 * 512 * 4);
  w.xyzA = (float*)alloc(BNmax * 3 * 4);
  w.xyzB = (float*)alloc(BNmax * 3 * 4);
  w.ptsA = (float*)alloc(BNmax * 32 * 4);                    // max rows*C = 2048*32
  w.ptsB = (float*)alloc(BNmax * 32 * 4);
  w.MEAN = (float*)alloc((size_t)BATCH * DM * 4);
  w.HD1  = (float*)alloc((size_t)BATCH * 256 * 4);
  w.HD2  = (float*)alloc((size_t)BATCH * 64 * 4);
  w.knn    = (int*)alloc(BNmax * (NNB + GCLS) * 4);
  w.fpsidx = (int*)alloc((size_t)BATCH * 512 * 4);

  // ---- xyz = x[..., :3] ----
  {
    size_t t = BNmax * 3;
    k_slice_xyz<<<nb(t, 256), 256, 0, stream>>>(x, w.xyzA, t);
  }
  // ---- points = mlp2(x, fc1): 6->32 relu -> 32 ----
  gemm(x,    fc1.l1.w, fc1.l1.b, w.H1,   BATCH * NPTS, 6,  32, true,  stream);
  gemm(w.H1, fc1.l2.w, fc1.l2.b, w.ptsA, BATCH * NPTS, 32, 32, false, stream);

  // ---- t1: transformer, no global attn ----
  run_tf(t1, w.xyzA, w.ptsA, w.ptsB, NPTS, 32, NNB, 0, w, stream);

  float* cxyz = w.xyzA;
  float* cpts = w.ptsB;
  int N = NPTS;

  for (int i = 0; i < 4; ++i) {
    const int ch  = 32 << (i + 1);        // 64,128,256,512
    const int Cin = ch / 2;
    const int Np  = NPTS >> (2 * (i + 1)); // 512,128,32,8

    run_sa(blk[i].td, cxyz, cpts, N, Cin, Np, ch, w, stream);

    // cls branch: rows 0..7 of old points -> mlp2(cls)
    {
      size_t t = (size_t)BATCH * GCLS * Cin;
      k_copy2d<<<nb(t, 256), 256, 0, stream>>>(cpts, w.CLSI, GCLS, Cin, N, GCLS, 0, 0, t);
    }
    gemm(w.CLSI, blk[i].td.cls.l1.w, blk[i].td.cls.l1.b, w.CLSH, BATCH * GCLS, Cin, ch, true,  stream);
    gemm(w.CLSH, blk[i].td.cls.l2.w, blk[i].td.cls.l2.b, w.CLSO, BATCH * GCLS, ch,  ch, false, stream);

    const int Nn = Np + GCLS;
    float* dxyz = (cxyz == w.xyzA) ? w.xyzB : w.xyzA;
    float* dpts = (cpts == w.ptsA) ? w.ptsB : w.ptsA;
    // xyz concat
    { size_t t = (size_t)BATCH * Np * 3;
      k_copy2d<<<nb(t, 256), 256, 0, stream>>>(w.NX, dxyz, Np, 3, Np, Nn, 0, 0, t); }
    { size_t t = (size_t)BATCH * GCLS * 3;
      k_copy2d<<<nb(t, 256), 256, 0, stream>>>(cxyz, dxyz, GCLS, 3, N, Nn, 0, Np, t); }
    // points concat
    { size_t t = (size_t)BATCH * Np * ch;
      k_copy2d<<<nb(t, 256), 256, 0, stream>>>(w.NP, dpts, Np, ch, Np, Nn, 0, 0, t); }
    { size_t t = (size_t)BATCH * GCLS * ch;
      k_copy2d<<<nb(t, 256), 256, 0, stream>>>(w.CLSO, dpts, GCLS, ch, GCLS, Nn, 0, Np, t); }

    // transformer with global attention; output into the now-free old pts buffer
    float* opts = cpts;
    run_tf(blk[i].tf, dxyz, dpts, opts, Nn, ch, NNB, GCLS, w, stream);

    cxyz = dxyz; cpts = opts; N = Nn;
  }

  // ---- head: mean over points -> 512->256 relu ->64 relu ->40 ----
  {
    size_t t = (size_t)BATCH * DM;
    k_mean<<<nb(t, 256), 256, 0, stream>>>(cpts, w.MEAN, N, DM, t);
  }
  float* out = (float*)d_out;
  gemm(w.MEAN, hl1.w, hl1.b, w.HD1, BATCH, DM,  256, true,  stream);
  gemm(w.HD1,  hl2.w, hl2.b, w.HD2, BATCH, 256, 64,  true,  stream);
  gemm(w.HD2,  hl3.w, hl3.b, out,   BATCH, 64,  40,  false, stream);
}